// Generator_50637664420120
// MI455X (gfx1250) — compile-verified
//
#include <hip/hip_runtime.h>
#include <math.h>

#define KSPLIT 8
#define EPS 1e-3f

typedef float v2f __attribute__((ext_vector_type(2)));
typedef float v8f __attribute__((ext_vector_type(8)));

// ---------------------------------------------------------------------------
// WMMA-based GEMV partial: y_part[g, j] = sum_{k in chunk g} x[k] * W[k*N + j]
// One wave (32 threads) per 16-column tile per k-chunk.
// A (16x4, f32): replicated x along M. Lane layout per ISA:
//   v0: lanes0-15 -> K=0, lanes16-31 -> K=2 ; v1: K=1 / K=3
// B (4x16, f32): natural W tile rows:
//   v0: lanes0-15 -> row k+0 (N=lane), lanes16-31 -> row k+2 (N=lane-16)
//   v1: rows k+1 / k+3
// Result C has all 16 rows identical; lane n (n<16) holds y[jbase+n] in c[0].
// ---------------------------------------------------------------------------
__global__ __launch_bounds__(32)
void gemv_wmma_part(const float* __restrict__ x, const float* __restrict__ W,
                    float* __restrict__ part, int K, int N)
{
    const int lane  = threadIdx.x;
    const int jbase = blockIdx.x * 16;
    const int g     = blockIdx.y;
    const int kchunk = K / KSPLIT;           // multiple of 4 for all layers here
    const int k0     = g * kchunk;
    const bool hi    = lane >= 16;
    const int  jl    = hi ? (lane - 16) : lane;

    v8f c = {};
    for (int k = k0; k < k0 + kchunk; k += 4) {
        v2f a;
        a.x = hi ? x[k + 2] : x[k + 0];
        a.y = hi ? x[k + 3] : x[k + 1];
        const float* Wr = W + (size_t)(k + (hi ? 2 : 0)) * N + jbase + jl;
        v2f b;
        b.x = Wr[0];
        b.y = Wr[N];
        c = __builtin_amdgcn_wmma_f32_16x16x4_f32(
                false, a, false, b, (short)0, c, false, false);
    }
    if (lane < 16)
        part[(size_t)g * N + jbase + lane] = c[0];
}

// y[j] = act( bias[j] + sum_g part[g*N + j] )
__global__ void gemv_finish(const float* __restrict__ part,
                            const float* __restrict__ bias,
                            float* __restrict__ y, int N, int relu)
{
    int j = blockIdx.x * blockDim.x + threadIdx.x;
    if (j >= N) return;
    float s = bias[j];
    for (int g = 0; g < KSPLIT; ++g) s += part[(size_t)g * N + j];
    if (relu) s = fmaxf(s, 0.0f);
    y[j] = s;
}

// Two-pass LayerNorm over n elements (single block).
__global__ __launch_bounds__(256)
void ln_kernel(const float* __restrict__ x, const float* __restrict__ g,
               const float* __restrict__ b, float* __restrict__ y, int n)
{
    __shared__ float red[256];
    const int t = threadIdx.x;
    float s = 0.0f;
    for (int i = t; i < n; i += 256) s += x[i];
    red[t] = s; __syncthreads();
    for (int o = 128; o > 0; o >>= 1) { if (t < o) red[t] += red[t + o]; __syncthreads(); }
    const float mean = red[0] / (float)n;
    __syncthreads();
    float s2 = 0.0f;
    for (int i = t; i < n; i += 256) { float d = x[i] - mean; s2 += d * d; }
    red[t] = s2; __syncthreads();
    for (int o = 128; o > 0; o >>= 1) { if (t < o) red[t] += red[t + o]; __syncthreads(); }
    const float inv = rsqrtf(red[0] / (float)n + EPS);
    for (int i = t; i < n; i += 256)
        y[i] = (x[i] - mean) * inv * g[i] + b[i];
}

// Block0 V row: y[j] = ln1_b[0] * wv[j] + bv[j]   (LN over size-1 axis == bias)
__global__ void vrow0_kernel(const float* __restrict__ ln1b,
                             const float* __restrict__ wv,
                             const float* __restrict__ bv,
                             float* __restrict__ y, int n)
{
    int j = blockIdx.x * blockDim.x + threadIdx.x;
    if (j < n) y[j] = ln1b[0] * wv[j] + bv[j];
}

__global__ void add_kernel(const float* __restrict__ a, const float* __restrict__ b,
                           float* __restrict__ y, int n)
{
    int j = blockIdx.x * blockDim.x + threadIdx.x;
    if (j < n) y[j] = a[j] + b[j];
}

// Partial sums of sum_k (a[k>>11] + u[k&2047] + u1[k&2047]) * w8[k]
__global__ __launch_bounds__(256)
void bigdot_part(const float* __restrict__ aS, const float* __restrict__ u,
                 const float* __restrict__ u1, const float* __restrict__ w8,
                 float* __restrict__ bpart)
{
    __shared__ float red[256];
    const int t = threadIdx.x;
    const size_t total = (size_t)2048 * 2048;
    float s = 0.0f;
    for (size_t k = (size_t)blockIdx.x * 256 + t; k < total;
         k += (size_t)gridDim.x * 256) {
        int si = (int)(k >> 11);
        int j  = (int)(k & 2047);
        s += (aS[si] + u[j] + u1[j]) * w8[k];
    }
    red[t] = s; __syncthreads();
    for (int o = 128; o > 0; o >>= 1) { if (t < o) red[t] += red[t + o]; __syncthreads(); }
    if (t == 0) bpart[blockIdx.x] = red[0];
}

__global__ __launch_bounds__(256)
void final_sigmoid(const float* __restrict__ bpart, const float* __restrict__ b8,
                   float* __restrict__ out, int nb)
{
    __shared__ float red[256];
    const int t = threadIdx.x;
    red[t] = (t < nb) ? bpart[t] : 0.0f;
    __syncthreads();
    for (int o = 128; o > 0; o >>= 1) { if (t < o) red[t] += red[t + o]; __syncthreads(); }
    if (t == 0) {
        float v = red[0] + b8[0];
        out[0] = 1.0f / (1.0f + expf(-v));
    }
}

// ---------------------------------------------------------------------------
extern "C" void kernel_launch(void* const* d_in, const int* in_sizes, int n_in,
                              void* d_out, int out_size, void* d_ws, size_t ws_size,
                              hipStream_t stream)
{
    (void)in_sizes; (void)n_in; (void)out_size; (void)ws_size;
    const float* z = (const float*)d_in[0];
    // MLP chain weights: indices 1..14 (w1,b1,...,w7,b7)
    const float* w[8]; const float* wb[8];
    for (int i = 0; i < 7; ++i) {
        w[i + 1]  = (const float*)d_in[1 + 2 * i];
        wb[i + 1] = (const float*)d_in[2 + 2 * i];
    }
    // Block layout (setup_inputs dict order):
    // ln1_g, ln1_b, wq, wk, wv, bq, bk, bv, bo, wo, ln2_g, ln2_b, fw1, fb1, fw2, fb2
    const int B0 = 15, B1 = 31;
    const float* b0_ln1_b = (const float*)d_in[B0 + 1];
    const float* b0_wv    = (const float*)d_in[B0 + 4];
    const float* b0_bv    = (const float*)d_in[B0 + 7];
    const float* b0_bo    = (const float*)d_in[B0 + 8];
    const float* b0_wo    = (const float*)d_in[B0 + 9];
    const float* b0_ln2_g = (const float*)d_in[B0 + 10];
    const float* b0_ln2_b = (const float*)d_in[B0 + 11];
    const float* b0_fw1   = (const float*)d_in[B0 + 12];
    const float* b0_fb1   = (const float*)d_in[B0 + 13];
    const float* b0_fw2   = (const float*)d_in[B0 + 14];
    const float* b0_fb2   = (const float*)d_in[B0 + 15];

    const float* b1_ln1_g = (const float*)d_in[B1 + 0];
    const float* b1_ln1_b = (const float*)d_in[B1 + 1];
    const float* b1_wv    = (const float*)d_in[B1 + 4];
    const float* b1_bv    = (const float*)d_in[B1 + 7];
    const float* b1_bo    = (const float*)d_in[B1 + 8];
    const float* b1_wo    = (const float*)d_in[B1 + 9];
    const float* b1_ln2_g = (const float*)d_in[B1 + 10];
    const float* b1_ln2_b = (const float*)d_in[B1 + 11];
    const float* b1_fw1   = (const float*)d_in[B1 + 12];
    const float* b1_fb1   = (const float*)d_in[B1 + 13];
    const float* b1_fw2   = (const float*)d_in[B1 + 14];
    const float* b1_fb2   = (const float*)d_in[B1 + 15];

    const float* w8 = (const float*)d_in[47];
    const float* b8 = (const float*)d_in[48];

    // Workspace layout (floats)
    float* wsf   = (float*)d_ws;
    float* bufA  = wsf;             // 2048
    float* bufB  = wsf + 2048;      // 2048
    float* aS    = wsf + 4096;      // 2048  (MLP output, "a[s]")
    float* uB    = wsf + 6144;      // 2048  (block0 FFN row)
    float* u1B   = wsf + 8192;      // 2048  (block1 FFN row)
    float* tB    = wsf + 10240;     // 2048
    float* part  = wsf + 12288;     // KSPLIT*2048
    float* bpart = wsf + 12288 + KSPLIT * 2048; // 256

    auto gemv = [&](const float* x, const float* W, const float* bias,
                    float* y, int K, int N, int relu) {
        dim3 grid(N / 16, KSPLIT);
        gemv_wmma_part<<<grid, 32, 0, stream>>>(x, W, part, K, N);
        gemv_finish<<<(N + 255) / 256, 256, 0, stream>>>(part, bias, y, N, relu);
    };

    // ---- MLP chain: 128 -> 32 -> 64 -> 128 -> 256 -> 512 -> 1024 -> 2048 ----
    gemv(z,    w[1], wb[1], bufA,  128,   32, 1);
    gemv(bufA, w[2], wb[2], bufB,   32,   64, 1);
    gemv(bufB, w[3], wb[3], bufA,   64,  128, 1);
    gemv(bufA, w[4], wb[4], bufB,  128,  256, 1);
    gemv(bufB, w[5], wb[5], bufA,  256,  512, 1);
    gemv(bufA, w[6], wb[6], bufB,  512, 1024, 1);
    gemv(bufB, w[7], wb[7], aS,   1024, 2048, 1);

    // ---- Block 0 (collapsed): MHA == ((c*wv)+bv) @ wo + bo, one row ----
    vrow0_kernel<<<8, 256, 0, stream>>>(b0_ln1_b, b0_wv, b0_bv, bufA, 2048);
    gemv(bufA, b0_wo, b0_bo, bufB, 2048, 2048, 0);                 // r0
    ln_kernel<<<1, 256, 0, stream>>>(bufB, b0_ln2_g, b0_ln2_b, bufA, 2048); // n0
    gemv(bufA, b0_fw1, b0_fb1, bufB, 2048, 2048, 1);
    gemv(bufB, b0_fw2, b0_fb2, uB,   2048, 2048, 1);               // u

    // ---- Block 1 (collapsed): all rows identical again ----
    ln_kernel<<<1, 256, 0, stream>>>(uB, b1_ln1_g, b1_ln1_b, bufA, 2048);   // h
    gemv(bufA, b1_wv, b1_bv, bufB, 2048, 2048, 0);                 // v row
    gemv(bufB, b1_wo, b1_bo, bufA, 2048, 2048, 0);                 // r1
    add_kernel<<<8, 256, 0, stream>>>(uB, bufA, tB, 2048);         // t = u + r1
    ln_kernel<<<1, 256, 0, stream>>>(tB, b1_ln2_g, b1_ln2_b, bufA, 2048);   // n1
    gemv(bufA, b1_fw1, b1_fb1, bufB, 2048, 2048, 1);
    gemv(bufB, b1_fw2, b1_fb2, u1B,  2048, 2048, 1);               // u1

    // ---- Final: sigmoid( sum_k (a[s]+u[j]+u1[j]) * w8[k] + b8 ) ----
    bigdot_part<<<256, 256, 0, stream>>>(aS, uB, u1B, w8, bpart);
    final_sigmoid<<<1, 256, 0, stream>>>(bpart, b8, (float*)d_out, 256);
}